// ConditionalLayer_76055280877934
// MI455X (gfx1250) — compile-verified
//
#include <hip/hip_runtime.h>

// Problem sizes (fixed by the reference)
#define B_SZ 8192
#define D_SZ 1024
#define C_SZ 8
#define PADB (B_SZ + C_SZ * 32)   // segments padded to 32 -> capacity 8448
#define SUPERTILES (PADB / 32)    // 264 row supertiles of 32
#define NBLK (D_SZ / 64)          // 16 n-blocks of width 64

typedef __attribute__((ext_vector_type(2))) float v2f;
typedef __attribute__((ext_vector_type(8))) float v8f;

// ---------------- pass 1: group rows by condition ----------------

__global__ void k_count(const int* __restrict__ cid, int* __restrict__ counts) {
  int i = blockIdx.x * blockDim.x + threadIdx.x;
  if (i < B_SZ) atomicAdd(&counts[cid[i]], 1);
}

__global__ void k_offsets(const int* __restrict__ counts, int* __restrict__ offs) {
  // single thread: tiny C=8 prefix sum, each segment start 32-aligned
  if (threadIdx.x == 0 && blockIdx.x == 0) {
    int acc = 0;
    for (int c = 0; c < C_SZ; ++c) {
      offs[c] = acc;
      acc += ((counts[c] + 31) >> 5) << 5;
    }
    offs[C_SZ] = acc;
  }
}

__global__ void k_scatter(const int* __restrict__ cid, const int* __restrict__ offs,
                          int* __restrict__ cursor, int* __restrict__ perm) {
  int i = blockIdx.x * blockDim.x + threadIdx.x;
  if (i < B_SZ) {
    int c = cid[i];
    int pos = offs[c] + atomicAdd(&cursor[c], 1);
    perm[pos] = i;
  }
}

__global__ void k_bias(const float* __restrict__ b, float* __restrict__ bias_sum) {
  int i = blockIdx.x * blockDim.x + threadIdx.x;
  if (i < D_SZ) {
    float s = 0.f;
#pragma unroll
    for (int c = 0; c < C_SZ; ++c) s += b[c * D_SZ + i];
    bias_sum[i] = s;
  }
}

// ---------------- pass 2: grouped GEMM via V_WMMA_F32_16X16X4_F32 ----------------
// One wave computes a 32(M) x 64(N) output strip: 2 M-subtiles x 4 N-subtiles.
// Per K-step (K advances 4): 2 A-fragment loads + 4 B-fragment loads feed 8 WMMAs
// (each B fragment reused across both M-subtiles) -> ~10.7 FLOP/byte from L2.
//
// fp32 WMMA operand layout (ISA 7.12.2, 32-bit 16x4 A / 4x16 B):
//   A: lanes 0-15 hold M=lane, {VGPR0:K=k0, VGPR1:K=k0+1}; lanes 16-31 hold K=k0+2,k0+3
//   B: lanes 0-15 hold N=lane, same K split
//   C/D: lane l holds N=l&15, VGPR v holds M = v + 8*(l>>4)
__global__ __launch_bounds__(32)
void k_gemm(const float* __restrict__ x, const int* __restrict__ cid,
            const float* __restrict__ W, const float* __restrict__ bias_sum,
            const int* __restrict__ perm, float* __restrict__ out) {
  const int lane = threadIdx.x;       // 0..31 (wave32)
  const int m    = lane & 15;
  const int hi   = lane >> 4;         // 0: K pair {0,1}; 1: K pair {2,3}
  const int st   = blockIdx.x;        // 32-row supertile in permuted order
  const int nb   = blockIdx.y;        // 64-wide N block
  const int nbase = nb * 64;

  // Row indices for this lane's two M-subtile slots (-1 = padding slot)
  const int r0 = perm[st * 32 + m];
  const int r1 = perm[st * 32 + 16 + m];

  // Wave-uniform condition id: min over valid lanes (supertile is condition-uniform)
  int c = (r0 >= 0) ? cid[r0] : 0x7FFFFFFF;
  if (r1 >= 0) { int c1 = cid[r1]; c = (c1 < c) ? c1 : c; }
#pragma unroll
  for (int off = 16; off > 0; off >>= 1) {
    int o = __shfl_xor(c, off, 32);
    c = (o < c) ? o : c;
  }
  if (c == 0x7FFFFFFF) return;  // entirely padded supertile (uniform branch)

  const float vf0 = (r0 >= 0) ? 1.0f : 0.0f;   // zero padding rows of A
  const float vf1 = (r1 >= 0) ? 1.0f : 0.0f;
  const float* xrow0 = x + (size_t)((r0 >= 0) ? r0 : 0) * D_SZ;
  const float* xrow1 = x + (size_t)((r1 >= 0) ? r1 : 0) * D_SZ;

  // B[k,n] = W[c, n, k]: per lane n is fixed -> contiguous K along last dim
  const float* wbase = W + (size_t)c * D_SZ * D_SZ;
  const float* wb0 = wbase + (size_t)(nbase +  0 + m) * D_SZ;
  const float* wb1 = wbase + (size_t)(nbase + 16 + m) * D_SZ;
  const float* wb2 = wbase + (size_t)(nbase + 32 + m) * D_SZ;
  const float* wb3 = wbase + (size_t)(nbase + 48 + m) * D_SZ;

  v8f a00 = {}, a01 = {}, a02 = {}, a03 = {};   // M-subtile 0 accumulators
  v8f a10 = {}, a11 = {}, a12 = {}, a13 = {};   // M-subtile 1 accumulators
  const int koff = hi * 2;

#pragma unroll 4
  for (int k0 = 0; k0 < D_SZ; k0 += 4) {
    v2f a0 = *(const v2f*)(xrow0 + k0 + koff);  a0 *= vf0;
    v2f a1 = *(const v2f*)(xrow1 + k0 + koff);  a1 *= vf1;
    v2f b0 = *(const v2f*)(wb0 + k0 + koff);
    v2f b1 = *(const v2f*)(wb1 + k0 + koff);
    v2f b2 = *(const v2f*)(wb2 + k0 + koff);
    v2f b3 = *(const v2f*)(wb3 + k0 + koff);
    // 8 args: (neg_a, A, neg_b, B, c_mod, C, reuse_a, reuse_b)
    a00 = __builtin_amdgcn_wmma_f32_16x16x4_f32(false, a0, false, b0, (short)0, a00, false, false);
    a01 = __builtin_amdgcn_wmma_f32_16x16x4_f32(false, a0, false, b1, (short)0, a01, false, false);
    a02 = __builtin_amdgcn_wmma_f32_16x16x4_f32(false, a0, false, b2, (short)0, a02, false, false);
    a03 = __builtin_amdgcn_wmma_f32_16x16x4_f32(false, a0, false, b3, (short)0, a03, false, false);
    a10 = __builtin_amdgcn_wmma_f32_16x16x4_f32(false, a1, false, b0, (short)0, a10, false, false);
    a11 = __builtin_amdgcn_wmma_f32_16x16x4_f32(false, a1, false, b1, (short)0, a11, false, false);
    a12 = __builtin_amdgcn_wmma_f32_16x16x4_f32(false, a1, false, b2, (short)0, a12, false, false);
    a13 = __builtin_amdgcn_wmma_f32_16x16x4_f32(false, a1, false, b3, (short)0, a13, false, false);
  }

  // Shared bias vector (added to every row per the reference's affine-sum identity)
  const float bias0 = bias_sum[nbase +  0 + m];
  const float bias1 = bias_sum[nbase + 16 + m];
  const float bias2 = bias_sum[nbase + 32 + m];
  const float bias3 = bias_sum[nbase + 48 + m];

  // Scatter D tiles: VGPR v corresponds to tile row v + 8*hi -> fetch perm via shuffle
#pragma unroll
  for (int v = 0; v < 8; ++v) {
    const int src = v + hi * 8;
    const int rr0 = __shfl(r0, src, 32);
    const int rr1 = __shfl(r1, src, 32);
    if (rr0 >= 0) {
      float* orow = out + (size_t)rr0 * D_SZ + nbase + m;
      orow[ 0] = a00[v] + bias0;
      orow[16] = a01[v] + bias1;
      orow[32] = a02[v] + bias2;
      orow[48] = a03[v] + bias3;
    }
    if (rr1 >= 0) {
      float* orow = out + (size_t)rr1 * D_SZ + nbase + m;
      orow[ 0] = a10[v] + bias0;
      orow[16] = a11[v] + bias1;
      orow[32] = a12[v] + bias2;
      orow[48] = a13[v] + bias3;
    }
  }
}

// ---------------- launch ----------------
// Workspace layout (bytes):
//   [0,32)      counts[8]
//   [64,96)     cursor[8]
//   [128,164)   offs[9]
//   [256,4352)  bias_sum[1024]
//   [4352, 4352+4*PADB) perm[8448]      -> total ~38.1 KB
extern "C" void kernel_launch(void* const* d_in, const int* in_sizes, int n_in,
                              void* d_out, int out_size, void* d_ws, size_t ws_size,
                              hipStream_t stream) {
  const float* x   = (const float*)d_in[0];
  const int*   cid = (const int*)d_in[1];
  const float* W   = (const float*)d_in[2];
  const float* b   = (const float*)d_in[3];
  float*       out = (float*)d_out;

  char* ws = (char*)d_ws;
  int*   counts   = (int*)(ws + 0);
  int*   cursor   = (int*)(ws + 64);
  int*   offs     = (int*)(ws + 128);
  float* bias_sum = (float*)(ws + 256);
  int*   perm     = (int*)(ws + 4352);

  hipMemsetAsync(counts, 0, 128, stream);                   // counts + cursor
  hipMemsetAsync(perm, 0xFF, PADB * sizeof(int), stream);   // all slots = -1

  k_count  <<<(B_SZ + 255) / 256, 256, 0, stream>>>(cid, counts);
  k_offsets<<<1, 1, 0, stream>>>(counts, offs);
  k_scatter<<<(B_SZ + 255) / 256, 256, 0, stream>>>(cid, offs, cursor, perm);
  k_bias   <<<(D_SZ + 255) / 256, 256, 0, stream>>>(b, bias_sum);

  dim3 grid(SUPERTILES, NBLK);
  k_gemm<<<grid, 32, 0, stream>>>(x, cid, W, bias_sum, perm, out);
}